// GAT_26877905338555
// MI455X (gfx1250) — compile-verified
//
#include <hip/hip_runtime.h>
#include <hip/hip_bf16.h>
#include <math.h>

typedef float v2f __attribute__((ext_vector_type(2)));
typedef float v8f __attribute__((ext_vector_type(8)));

#define GAT_INC 256
#define GAT_HID 64
#define NEG_SLOPE 0.2f

// ---------------------------------------------------------------------------
// GEMM: C[N x 64] = A[N x K] @ W[K x 64], f32 WMMA 16x16x4.
// Block = 256 threads (8 waves), M-tile = 64 rows.
// Wave w owns a 32-row x 16-col sub-tile: rows (w>>2)*32 .. +31 (two 16-row
// WMMA accumulators), cols (w&3)*16 .. +15. B registers are loaded once per
// k-step and reused by both WMMAs. A-tile staged in LDS with padded stride.
// ---------------------------------------------------------------------------
__global__ __launch_bounds__(256)
void gat_gemm_wmma(const float* __restrict__ A, const float* __restrict__ W,
                   float* __restrict__ C, int N, int K) {
    __shared__ float sA[64 * (GAT_INC + 4)];   // max K = 256, pad 4 floats
    const int KP   = K + 4;
    const int lane = threadIdx.x & 31;
    const int wave = threadIdx.x >> 5;
    const int rowBase = blockIdx.x * 64;

    const int col      = (wave & 3) * 16 + (lane & 15);
    const int waveRow  = (wave >> 2) * 32;     // 0 or 32

    // Warm L2/WGP$ for the B panel (global_prefetch_b8).
    for (int k = threadIdx.x; k < K; k += 256)
        __builtin_prefetch(&W[(size_t)k * 64 + col], 0, 3);

    // Coalesced fill of 64 x K A-tile.
    for (int idx = threadIdx.x; idx < 64 * K; idx += 256) {
        int r = idx / K, c = idx - r * K;
        int gr = rowBase + r;
        sA[r * KP + c] = (gr < N) ? A[(size_t)gr * K + c] : 0.0f;
    }
    __syncthreads();

    const int row   = lane & 15;           // A row within 16-row sub-tile
    const int khalf = (lane >> 4) << 1;    // lanes 0-15 -> K 0/1, 16-31 -> K 2/3

    v8f acc0 = {};
    v8f acc1 = {};
    for (int k = 0; k < K; k += 4) {
        v2f b;
        b.x = W[(size_t)(k + khalf)     * 64 + col];
        b.y = W[(size_t)(k + khalf + 1) * 64 + col];

        v2f a0, a1;
        a0.x = sA[(waveRow +      row) * KP + k + khalf];
        a0.y = sA[(waveRow +      row) * KP + k + khalf + 1];
        a1.x = sA[(waveRow + 16 + row) * KP + k + khalf];
        a1.y = sA[(waveRow + 16 + row) * KP + k + khalf + 1];

        acc0 = __builtin_amdgcn_wmma_f32_16x16x4_f32(
            false, a0, false, b, (short)0, acc0, false, false);
        acc1 = __builtin_amdgcn_wmma_f32_16x16x4_f32(
            false, a1, false, b, (short)0, acc1, false, false);
    }

    // D layout: VGPR r -> M=r (lanes 0-15) / M=r+8 (lanes 16-31), N=lane&15
    const int mhalf = (lane >> 4) << 3;
    for (int r = 0; r < 8; ++r) {
        int gr0 = rowBase + waveRow +      mhalf + r;
        int gr1 = rowBase + waveRow + 16 + mhalf + r;
        if (gr0 < N) C[(size_t)gr0 * 64 + col] = acc0[r];
        if (gr1 < N) C[(size_t)gr1 * 64 + col] = acc1[r];
    }
}

// ---------------------------------------------------------------------------
// Per-node attention logits: as_[n] = h[n,:].a_src ; ad_[n] = h[n,:].a_dst
// One wave32 per node, 2 features per lane, shuffle reduction.
// ---------------------------------------------------------------------------
__global__ __launch_bounds__(256)
void gat_node_alpha(const float* __restrict__ h, const float* __restrict__ a_src,
                    const float* __restrict__ a_dst, float* __restrict__ as_,
                    float* __restrict__ ad_, int N) {
    int node = (int)((blockIdx.x * (size_t)blockDim.x + threadIdx.x) >> 5);
    int lane = threadIdx.x & 31;
    if (node >= N) return;
    const float* hr = h + (size_t)node * 64;
    float s = hr[lane] * a_src[lane] + hr[lane + 32] * a_src[lane + 32];
    float d = hr[lane] * a_dst[lane] + hr[lane + 32] * a_dst[lane + 32];
    for (int off = 16; off > 0; off >>= 1) {
        s += __shfl_down(s, off, 32);
        d += __shfl_down(d, off, 32);
    }
    if (lane == 0) { as_[node] = s; ad_[node] = d; }
}

// ---------------------------------------------------------------------------
// Init: zero agg[N*64], emax = -inf-ish, denom = 0.
// ---------------------------------------------------------------------------
__global__ __launch_bounds__(256)
void gat_init(float* __restrict__ agg, float* __restrict__ emax,
              float* __restrict__ denom, int N) {
    int i = (int)(blockIdx.x * (size_t)blockDim.x + threadIdx.x);
    if (i < N * 64) agg[i] = 0.0f;
    if (i < N) { emax[i] = -3.0e38f; denom[i] = 0.0f; }
}

// ---------------------------------------------------------------------------
// Edge pass 1: e = leaky_relu(as[src] + ad[dst]); segment max via f32 atomics.
// ---------------------------------------------------------------------------
__global__ __launch_bounds__(256)
void gat_edge_max(const int* __restrict__ srcI, const int* __restrict__ dstI,
                  const float* __restrict__ as_, const float* __restrict__ ad_,
                  float* __restrict__ ebuf, float* __restrict__ emax, int E) {
    int e = (int)(blockIdx.x * (size_t)blockDim.x + threadIdx.x);
    if (e >= E) return;
    int s = srcI[e], d = dstI[e];
    float v = as_[s] + ad_[d];
    v = (v > 0.0f) ? v : v * NEG_SLOPE;
    ebuf[e] = v;
    atomicMax(&emax[d], v);              // global_atomic_max_num_f32
}

// ---------------------------------------------------------------------------
// Edge pass 2: ex = exp(e - emax[dst]); segment sum into denom.
// ---------------------------------------------------------------------------
__global__ __launch_bounds__(256)
void gat_edge_exp(const int* __restrict__ dstI, const float* __restrict__ emax,
                  float* __restrict__ ebuf, float* __restrict__ denom, int E) {
    int e = (int)(blockIdx.x * (size_t)blockDim.x + threadIdx.x);
    if (e >= E) return;
    int d = dstI[e];
    float ex = __expf(ebuf[e] - emax[d]);
    ebuf[e] = ex;
    atomicAdd(&denom[d], ex);
}

// ---------------------------------------------------------------------------
// Edge pass 3: agg[dst,:] += alpha * h[src,:]. One wave per edge, 2 feats/lane.
// ---------------------------------------------------------------------------
__global__ __launch_bounds__(256)
void gat_edge_aggr(const int* __restrict__ srcI, const int* __restrict__ dstI,
                   const float* __restrict__ ebuf, const float* __restrict__ denom,
                   const float* __restrict__ h, float* __restrict__ agg, int E) {
    int edge = (int)((blockIdx.x * (size_t)blockDim.x + threadIdx.x) >> 5);
    int lane = threadIdx.x & 31;
    if (edge >= E) return;
    int s = srcI[edge], d = dstI[edge];
    float alpha = ebuf[edge] / (denom[d] + 1e-16f);
    const float* hs = h + (size_t)s * 64;
    float*       od = agg + (size_t)d * 64;
    atomicAdd(&od[lane],      alpha * hs[lane]);
    atomicAdd(&od[lane + 32], alpha * hs[lane + 32]);
}

// ---------------------------------------------------------------------------
// Finalize: out = act(agg + bias). relu_flag selects ReLU (layer 1).
// ---------------------------------------------------------------------------
__global__ __launch_bounds__(256)
void gat_finalize(const float* __restrict__ agg, const float* __restrict__ b,
                  float* __restrict__ out, int N, int relu_flag) {
    int i = (int)(blockIdx.x * (size_t)blockDim.x + threadIdx.x);
    if (i >= N * 64) return;
    float v = agg[i] + b[i & 63];
    out[i] = relu_flag ? fmaxf(v, 0.0f) : v;
}

// ---------------------------------------------------------------------------
extern "C" void kernel_launch(void* const* d_in, const int* in_sizes, int n_in,
                              void* d_out, int out_size, void* d_ws, size_t ws_size,
                              hipStream_t stream) {
    const float* x    = (const float*)d_in[0];
    const int*   eidx = (const int*)  d_in[1];   // [2, E] flattened: src then dst
    const float* W1   = (const float*)d_in[2];
    const float* a1s  = (const float*)d_in[3];
    const float* a1d  = (const float*)d_in[4];
    const float* b1   = (const float*)d_in[5];
    const float* W2   = (const float*)d_in[6];
    const float* a2s  = (const float*)d_in[7];
    const float* a2d  = (const float*)d_in[8];
    const float* b2   = (const float*)d_in[9];

    const int N = in_sizes[0] / GAT_INC;
    const int E = in_sizes[1] / 2;
    const int* srcI = eidx;
    const int* dstI = eidx + E;

    float* ws = (float*)d_ws;
    size_t o = 0;
    float* h     = ws + o; o += (size_t)N * 64;
    float* hpost = ws + o; o += (size_t)N * 64;
    float* agg   = ws + o; o += (size_t)N * 64;
    float* as_   = ws + o; o += N;
    float* ad_   = ws + o; o += N;
    float* emax  = ws + o; o += N;
    float* denom = ws + o; o += N;
    float* ebuf  = ws + o; o += E;

    const int gGemm  = (N + 63) / 64;
    const int gNode  = (int)(((size_t)N * 32 + 255) / 256);
    const int gInit  = (N * 64 + 255) / 256;
    const int gEdge  = (E + 255) / 256;
    const int gEAgg  = (int)(((size_t)E * 32 + 255) / 256);
    const int gElem  = (N * 64 + 255) / 256;

    // ---------------- Layer 1 (x -> hpost) ----------------
    gat_gemm_wmma <<<gGemm, 256, 0, stream>>>(x, W1, h, N, GAT_INC);
    gat_node_alpha<<<gNode, 256, 0, stream>>>(h, a1s, a1d, as_, ad_, N);
    gat_init      <<<gInit, 256, 0, stream>>>(agg, emax, denom, N);
    gat_edge_max  <<<gEdge, 256, 0, stream>>>(srcI, dstI, as_, ad_, ebuf, emax, E);
    gat_edge_exp  <<<gEdge, 256, 0, stream>>>(dstI, emax, ebuf, denom, E);
    gat_edge_aggr <<<gEAgg, 256, 0, stream>>>(srcI, dstI, ebuf, denom, h, agg, E);
    gat_finalize  <<<gElem, 256, 0, stream>>>(agg, b1, hpost, N, 1);

    // ---------------- Layer 2 (hpost -> d_out) ----------------
    float* out = (float*)d_out;
    gat_gemm_wmma <<<gGemm, 256, 0, stream>>>(hpost, W2, h, N, GAT_HID);
    gat_node_alpha<<<gNode, 256, 0, stream>>>(h, a2s, a2d, as_, ad_, N);
    gat_init      <<<gInit, 256, 0, stream>>>(out, emax, denom, N);
    gat_edge_max  <<<gEdge, 256, 0, stream>>>(srcI, dstI, as_, ad_, ebuf, emax, E);
    gat_edge_exp  <<<gEdge, 256, 0, stream>>>(dstI, emax, ebuf, denom, E);
    gat_edge_aggr <<<gEAgg, 256, 0, stream>>>(srcI, dstI, ebuf, denom, h, out, E);
    gat_finalize  <<<gElem, 256, 0, stream>>>(out, b2, out, N, 0);
}